// BDEqualizer_79362405696081
// MI455X (gfx1250) — compile-verified
//
#include <hip/hip_runtime.h>
#include <stdint.h>

// Problem constants from the reference: B, NRX, NR, U, A, S, F = 4,1,32,4,8,14,1024
#define B_  4
#define NR_ 32
#define U_  4
#define A_  8
#define S_  14
#define F_  1024
#define TPB 256

typedef float v2f __attribute__((ext_vector_type(2)));

__device__ __forceinline__ void cswap(bool sw, float& a, float& b) {
    float t = a;
    a = sw ? b : a;
    b = sw ? t : b;
}

// One thread solves one 8x8 complex system H x = y for (b, s, f, u).
__global__ __launch_bounds__(TPB) void bdeq_solve_kernel(
    const float* __restrict__ y_re, const float* __restrict__ y_im,
    const float* __restrict__ h_re, const float* __restrict__ h_im,
    float* __restrict__ out)
{
    const int tid = threadIdx.x;
    const int f   = blockIdx.x * TPB + tid;   // gridDim.x = F/TPB = 4
    const int s   = blockIdx.y;               // gridDim.y = S = 14
    const int bz  = blockIdx.z;               // gridDim.z = B*U = 16
    const int b   = bz >> 2;
    const int u   = bz & 3;

    // ------------------------------------------------------------------
    // Stage y (rhs) through LDS using gfx1250 async global->LDS loads.
    // Each lane deposits its own 16 dwords (8 complex) and reads them back;
    // ordering is enforced with s_wait_asynccnt (no cross-lane sharing, so
    // no workgroup barrier is required).
    // ------------------------------------------------------------------
    __shared__ float sy[2 * A_][TPB];   // 16 KB

#pragma unroll
    for (int a1 = 0; a1 < A_; ++a1) {
        const int nr = u * A_ + a1;
        const size_t yoff = (((size_t)b * NR_ + nr) * S_ + s) * F_ + f;
        uint32_t lds_re = (uint32_t)(uintptr_t)&sy[a1][tid];
        uint32_t lds_im = (uint32_t)(uintptr_t)&sy[A_ + a1][tid];
        uint64_t g_re = (uint64_t)(uintptr_t)(y_re + yoff);
        uint64_t g_im = (uint64_t)(uintptr_t)(y_im + yoff);
        asm volatile("global_load_async_to_lds_b32 %0, %1, off"
                     :: "v"(lds_re), "v"(g_re) : "memory");
        asm volatile("global_load_async_to_lds_b32 %0, %1, off"
                     :: "v"(lds_im), "v"(g_im) : "memory");
    }
#if __has_builtin(__builtin_amdgcn_s_wait_asynccnt)
    __builtin_amdgcn_s_wait_asynccnt(0);
#else
    asm volatile("s_wait_asynccnt 0x0" ::: "memory");
#endif

    float yr[A_], yi[A_];
#pragma unroll
    for (int a1 = 0; a1 < A_; ++a1) {
        yr[a1] = sy[a1][tid];
        yi[a1] = sy[A_ + a1][tid];
    }

    // ------------------------------------------------------------------
    // Load the 8x8 complex H into registers. h is 470 MB streamed once
    // (exceeds the 192 MB L2) -> non-temporal loads. Consecutive lanes
    // read consecutive f -> each load is one fully-coalesced 128B line.
    // ------------------------------------------------------------------
    float Hr[A_][A_], Hi[A_][A_];
#pragma unroll
    for (int a1 = 0; a1 < A_; ++a1) {
        const int nr = u * A_ + a1;
#pragma unroll
        for (int a2 = 0; a2 < A_; ++a2) {
            const size_t hoff =
                (((((size_t)b * NR_ + nr) * U_ + u) * A_ + a2) * S_ + s) * (size_t)F_ + f;
            Hr[a1][a2] = __builtin_nontemporal_load(h_re + hoff);
            Hi[a1][a2] = __builtin_nontemporal_load(h_im + hoff);
        }
    }

    // ------------------------------------------------------------------
    // In-register complex LU with branchless partial pivoting.
    // Fully unrolled so every array index is a compile-time constant
    // (arrays live in VGPRs, no scratch).
    // ------------------------------------------------------------------
#pragma unroll
    for (int k = 0; k < A_; ++k) {
        // partial pivot: bubble the max-|H[i][k]| row up to row k
#pragma unroll
        for (int i = k + 1; i < A_; ++i) {
            float mk = Hr[k][k] * Hr[k][k] + Hi[k][k] * Hi[k][k];
            float mi = Hr[i][k] * Hr[i][k] + Hi[i][k] * Hi[i][k];
            bool sw = mi > mk;
#pragma unroll
            for (int j = k; j < A_; ++j) {
                cswap(sw, Hr[k][j], Hr[i][j]);
                cswap(sw, Hi[k][j], Hi[i][j]);
            }
            cswap(sw, yr[k], yr[i]);
            cswap(sw, yi[k], yi[i]);
        }
        // eliminate below pivot: factor = H[i][k] / H[k][k]
        const float pr = Hr[k][k], pi = Hi[k][k];
        const float inv = 1.0f / (pr * pr + pi * pi);
        const float ipr =  pr * inv;
        const float ipi = -pi * inv;
#pragma unroll
        for (int i = k + 1; i < A_; ++i) {
            const float fr = Hr[i][k] * ipr - Hi[i][k] * ipi;
            const float fi = Hr[i][k] * ipi + Hi[i][k] * ipr;
#pragma unroll
            for (int j = k + 1; j < A_; ++j) {
                Hr[i][j] -= fr * Hr[k][j] - fi * Hi[k][j];
                Hi[i][j] -= fr * Hi[k][j] + fi * Hr[k][j];
            }
            yr[i] -= fr * yr[k] - fi * yi[k];
            yi[i] -= fr * yi[k] + fi * yr[k];
        }
    }

    // back substitution
    float xr[A_], xi[A_];
#pragma unroll
    for (int k = A_ - 1; k >= 0; --k) {
        float sr = yr[k], si = yi[k];
#pragma unroll
        for (int j = k + 1; j < A_; ++j) {
            sr -= Hr[k][j] * xr[j] - Hi[k][j] * xi[j];
            si -= Hr[k][j] * xi[j] + Hi[k][j] * xr[j];
        }
        const float pr = Hr[k][k], pi = Hi[k][k];
        const float inv = 1.0f / (pr * pr + pi * pi);
        const float ipr =  pr * inv;
        const float ipi = -pi * inv;
        xr[k] = sr * ipr - si * ipi;
        xi[k] = sr * ipi + si * ipr;
    }

    // ------------------------------------------------------------------
    // Store: out[b, 0, u*8+a, s, f, {re,im}] -- (re,im) pair per thread is
    // one 8B-aligned float2; consecutive lanes are contiguous -> coalesced.
    // Output is write-once streamed -> non-temporal.
    // ------------------------------------------------------------------
#pragma unroll
    for (int a = 0; a < A_; ++a) {
        const int nr = u * A_ + a;
        const size_t ooff = ((((size_t)b * NR_ + nr) * S_ + s) * (size_t)F_ + f) * 2;
        v2f v;
        v.x = xr[a];
        v.y = xi[a];
        __builtin_nontemporal_store(v, (v2f*)(out + ooff));
    }
}

extern "C" void kernel_launch(void* const* d_in, const int* in_sizes, int n_in,
                              void* d_out, int out_size, void* d_ws, size_t ws_size,
                              hipStream_t stream) {
    const float* y_re = (const float*)d_in[0];
    const float* y_im = (const float*)d_in[1];
    const float* h_re = (const float*)d_in[2];
    const float* h_im = (const float*)d_in[3];
    float* out = (float*)d_out;

    dim3 grid(F_ / TPB, S_, B_ * U_);   // 4 x 14 x 16 = 896 blocks of 256
    bdeq_solve_kernel<<<grid, TPB, 0, stream>>>(y_re, y_im, h_re, h_im, out);
}